// anomaly_anticipation_52003464020266
// MI455X (gfx1250) — compile-verified
//
#include <hip/hip_runtime.h>
#include <hip/hip_bf16.h>

typedef float v2f __attribute__((ext_vector_type(2)));
typedef float v8f __attribute__((ext_vector_type(8)));

#define NODES_CH_IN 128
#define H1 300
#define H2 100
#define H3 32
#define F1 16
#define NT 4   // N-tiles per wave: 16x64 output strip

// ---------------------------------------------------------------------------
__global__ void zero_kernel(float* __restrict__ p, long long n) {
  long long i = (long long)blockIdx.x * blockDim.x + threadIdx.x;
  if (i < n) p[i] = 0.0f;
}

// Degree accumulation over dst (self-loop +1 folded into dis kernel)
__global__ void deg_kernel(const long long* __restrict__ dst, float* __restrict__ deg, int E) {
  int e = blockIdx.x * blockDim.x + threadIdx.x;
  if (e < E) atomicAdd(&deg[(int)dst[e]], 1.0f);
}

__global__ void dis_kernel(const float* __restrict__ deg, float* __restrict__ dis, int n) {
  int i = blockIdx.x * blockDim.x + threadIdx.x;
  if (i < n) dis[i] = rsqrtf(deg[i] + 1.0f);  // +1 = self loop
}

// Per-edge symmetric norm: norm[e] = dis[src]*dis[dst]
__global__ void norm_kernel(const long long* __restrict__ src,
                            const long long* __restrict__ dst,
                            const float* __restrict__ dis,
                            float* __restrict__ norm, int E) {
  int e = blockIdx.x * blockDim.x + threadIdx.x;
  if (e < E) norm[e] = dis[(int)src[e]] * dis[(int)dst[e]];
}

// ---------------------------------------------------------------------------
// fp32 WMMA GEMM: C[M,N] = A[M,K] @ B[K,N].
// One wave computes a 16x64 strip (4 tiles along N) so each A fragment feeds
// 4 v_wmma_f32_16x16x4_f32 ops. M multiple of 16, K multiple of 4; N edge
// tiles handled with clamped+masked columns (no EXEC divergence around WMMA).
__global__ void gemm_wmma_f32(const float* __restrict__ A, const float* __restrict__ B,
                              float* __restrict__ C, int M, int K, int N, int groupsN) {
  int gtid = blockIdx.x * blockDim.x + threadIdx.x;
  int wave = gtid >> 5;          // wave32
  int lane = threadIdx.x & 31;
  int tilesM = M >> 4;
  if (wave >= tilesM * groupsN) return;   // wave-uniform exit
  int tm = wave / groupsN;
  int tg = wave - tm * groupsN;

  bool hi = lane >= 16;
  int rowA = tm * 16 + (lane & 15);
  int colBase = tg * 64 + (lane & 15);

  int   colc[NT];
  float cmask[NT];
#pragma unroll
  for (int t = 0; t < NT; ++t) {
    int col = colBase + t * 16;
    colc[t]  = (col < N) ? col : 0;
    cmask[t] = (col < N) ? 1.0f : 0.0f;
  }

  // lanes 0-15 hold K={k,k+1}; lanes 16-31 hold K={k+2,k+3} (ISA 16x4 layout)
  const float* __restrict__ Arow = A + (long long)rowA * K + (hi ? 2 : 0);
  v8f acc[NT] = {};

  for (int k = 0; k < K; k += 4) {
    v2f a = *(const v2f*)(Arow + k);          // 8B-aligned contiguous pair
    const float* __restrict__ Bk = B + (long long)(k + (hi ? 2 : 0)) * N;
#pragma unroll
    for (int t = 0; t < NT; ++t) {
      v2f b;
      b.x = Bk[colc[t]]     * cmask[t];
      b.y = Bk[N + colc[t]] * cmask[t];
      acc[t] = __builtin_amdgcn_wmma_f32_16x16x4_f32(false, a, false, b,
                                                     (short)0, acc[t], false, false);
    }
  }

  int baseRow = tm * 16 + (hi ? 8 : 0);
#pragma unroll
  for (int t = 0; t < NT; ++t) {
    int col = colBase + t * 16;
    if (col < N) {
#pragma unroll
      for (int v = 0; v < 8; ++v)
        C[(long long)(baseRow + v) * N + col] = acc[t][v];
    }
  }
}

// ---------------------------------------------------------------------------
// Self-loop fused init: agg[i,c] = h[i,c] * dis[i]^2  (replaces zero-fill)
__global__ void selfloop_init_kernel(float* __restrict__ agg, const float* __restrict__ h,
                                     const float* __restrict__ dis, int C, long long total) {
  long long idx = (long long)blockIdx.x * blockDim.x + threadIdx.x;
  if (idx >= total) return;
  int i = (int)(idx / C);
  float di = dis[i];
  agg[idx] = h[idx] * di * di;
}

// Edge scatter: agg[dst] += h[src] * norm[e]. One wave per edge; lanes stride
// channels -> coalesced gathers + L2-resident global_atomic_add_f32.
__global__ void scatter_kernel(const float* __restrict__ h,
                               const long long* __restrict__ src,
                               const long long* __restrict__ dst,
                               const float* __restrict__ norm,
                               float* __restrict__ agg, int C, int E) {
  int gtid = blockIdx.x * blockDim.x + threadIdx.x;
  int lane = gtid & 31;
  int wave = gtid >> 5;
  int nwaves = (gridDim.x * blockDim.x) >> 5;
  for (int e = wave; e < E; e += nwaves) {
    int s = (int)src[e];
    int d = (int)dst[e];
    float w = norm[e];
    const float* hs = h + (long long)s * C;
    float* ad = agg + (long long)d * C;
    for (int c = lane; c < C; c += 32)
      atomicAdd(&ad[c], hs[c] * w);
  }
}

// In-place epilogue: agg = relu(agg + bias)
__global__ void bias_relu_kernel(float* __restrict__ agg, const float* __restrict__ b,
                                 int C, long long total) {
  long long idx = (long long)blockIdx.x * blockDim.x + threadIdx.x;
  if (idx >= total) return;
  int c = (int)(idx % C);
  float v = agg[idx] + b[c];
  agg[idx] = v > 0.0f ? v : 0.0f;
}

// ---------------------------------------------------------------------------
// MLP head: out[i] = (x4[i,:] @ Wf1 + bf1) @ Wf2 + bf2   (no relu, per reference)
__global__ void head_kernel(const float* __restrict__ x,
                            const float* __restrict__ Wf1, const float* __restrict__ bf1,
                            const float* __restrict__ Wf2, const float* __restrict__ bf2,
                            float* __restrict__ out, int n) {
  int i = blockIdx.x * blockDim.x + threadIdx.x;
  if (i >= n) return;
  float xi[H3];
#pragma unroll
  for (int c = 0; c < H3; ++c) xi[c] = x[(long long)i * H3 + c];
  float r = bf2[0];
#pragma unroll
  for (int j = 0; j < F1; ++j) {
    float t = bf1[j];
#pragma unroll
    for (int c = 0; c < H3; ++c) t += xi[c] * Wf1[c * F1 + j];
    r += t * Wf2[j];
  }
  out[i] = r;
}

// ---------------------------------------------------------------------------
static inline long long alignup(long long v) { return (v + 255) & ~255LL; }

extern "C" void kernel_launch(void* const* d_in, const int* in_sizes, int n_in,
                              void* d_out, int out_size, void* d_ws, size_t ws_size,
                              hipStream_t stream) {
  const float*     x  = (const float*)d_in[0];
  const long long* ei = (const long long*)d_in[1];   // int64 edge_index [2,E]
  const float* W1 = (const float*)d_in[2];  const float* b1 = (const float*)d_in[3];
  const float* W2 = (const float*)d_in[4];  const float* b2 = (const float*)d_in[5];
  const float* W3 = (const float*)d_in[6];  const float* b3 = (const float*)d_in[7];
  const float* Wf1 = (const float*)d_in[8]; const float* bf1 = (const float*)d_in[9];
  const float* Wf2 = (const float*)d_in[10]; const float* bf2 = (const float*)d_in[11];
  float* out = (float*)d_out;

  const int N = in_sizes[0] / NODES_CH_IN;   // 50000
  const int E = in_sizes[1] / 2;             // 800000
  const long long* src = ei;
  const long long* dst = ei + E;

  // Workspace carve-up
  char* ws = (char*)d_ws;
  long long off = 0;
  float* deg    = (float*)(ws + off); off = alignup(off + (long long)N * 4);
  float* dis    = (float*)(ws + off); off = alignup(off + (long long)N * 4);
  float* norm   = (float*)(ws + off); off = alignup(off + (long long)E * 4);
  float* bufH   = (float*)(ws + off); off = alignup(off + (long long)N * H1 * 4);
  float* bufAgg = (float*)(ws + off); off = alignup(off + (long long)N * H1 * 4);
  (void)ws_size;

  const int TB = 256;
  auto blocks1d = [](long long n, int tb) { return (int)((n + tb - 1) / tb); };

  // ---- degree + normalization -------------------------------------------
  zero_kernel<<<blocks1d(N, TB), TB, 0, stream>>>(deg, N);
  deg_kernel<<<blocks1d(E, TB), TB, 0, stream>>>(dst, deg, E);
  dis_kernel<<<blocks1d(N, TB), TB, 0, stream>>>(deg, dis, N);
  norm_kernel<<<blocks1d(E, TB), TB, 0, stream>>>(src, dst, dis, norm, E);

  const int scatterBlocks = 4096;  // 32768 waves grid-striding 800K edges

  // ---- GCN layers --------------------------------------------------------
  struct Layer { const float* W; const float* b; int K; int C; };
  Layer layers[3] = { {W1, b1, NODES_CH_IN, H1}, {W2, b2, H1, H2}, {W3, b3, H2, H3} };

  const float* act = x;
  for (int l = 0; l < 3; ++l) {
    int K = layers[l].K, C = layers[l].C;
    int groupsN = (C + 63) / 64;
    long long totalWaves = (long long)(N / 16) * groupsN;
    // GEMM: bufH = act @ W   (4x v_wmma_f32_16x16x4_f32 per K-step per wave)
    gemm_wmma_f32<<<blocks1d(totalWaves * 32, TB), TB, 0, stream>>>(
        act, layers[l].W, bufH, N, K, C, groupsN);
    // Aggregate: init with self-loop term, scatter edges, bias+relu
    long long tot = (long long)N * C;
    selfloop_init_kernel<<<blocks1d(tot, TB), TB, 0, stream>>>(bufAgg, bufH, dis, C, tot);
    scatter_kernel<<<scatterBlocks, TB, 0, stream>>>(bufH, src, dst, norm, bufAgg, C, E);
    bias_relu_kernel<<<blocks1d(tot, TB), TB, 0, stream>>>(bufAgg, layers[l].b, C, tot);
    act = bufAgg;  // consumed by next GEMM before bufAgg is re-initialized
  }

  // ---- MLP head ----------------------------------------------------------
  head_kernel<<<blocks1d(N, TB), TB, 0, stream>>>(bufAgg, Wf1, bf1, Wf2, bf2, out, N);
}